// RankLoss_80908593922473
// MI455X (gfx1250) — compile-verified
//
#include <hip/hip_runtime.h>

typedef __attribute__((ext_vector_type(2))) float v2f;
typedef __attribute__((ext_vector_type(8))) float v8f;

#define TILE 16

__global__ void rankloss_init(float* ws_sum, unsigned int* ws_cnt) {
    *ws_sum = 0.0f;
    *ws_cnt = 0u;
}

__global__ __launch_bounds__(256)
void rankloss_wmma(const float* __restrict__ pred,
                   const int* __restrict__ rank,
                   float* __restrict__ ws_sum,
                   unsigned int* __restrict__ ws_cnt) {
    const int lane = threadIdx.x & 31;
    const int wave = threadIdx.x >> 5;
    const int col  = lane & 15;   // row index for A, column index for B/D
    const int hi   = lane >> 4;   // 0 = lanes 0-15, 1 = lanes 16-31

    const int i0     = (blockIdx.x * 8 + wave) * TILE;  // this wave's i-tile
    const int j_base = blockIdx.y * 1024;               // 64 j-tiles of 16

    // A (16x4 f32): A[m][0] = pred_i[m], A[m][1] = 1, A[m][2..3] = 0
    // lanes 0-15 hold K=0 (x) / K=1 (y); lanes 16-31 hold K=2 (x) / K=3 (y)
    v2f a;
    a.x = (hi == 0) ? pred[i0 + col] : 0.0f;
    a.y = (hi == 0) ? 1.0f           : 0.0f;

    // rank_i for the 8 D-rows this lane owns: M = v + 8*hi
    int ri[8];
#pragma unroll
    for (int v = 0; v < 8; ++v) ri[v] = rank[i0 + v + 8 * hi];

    float        sum = 0.0f;
    unsigned int cnt = 0u;

    for (int jt = 0; jt < 64; ++jt) {
        const int   j0 = j_base + jt * TILE;
        const float pj = pred[j0 + col];
        const int   rj = rank[j0 + col];

        // B (4x16 f32): row K=0 = ones, row K=1 = -pred_j, rows 2,3 = 0
        // VGPR0: lanes 0-15 = K=0, lanes 16-31 = K=1; VGPR1 = K=2/3 (zero)
        v2f b;
        b.x = (hi == 0) ? 1.0f : -pj;
        b.y = 0.0f;

        v8f c;
#pragma unroll
        for (int v = 0; v < 8; ++v) c[v] = 1.0f;  // the "+1" bias

        // D[m][n] = 1 + pred_i[m] - pred_j[n], exact f32
        v8f d = __builtin_amdgcn_wmma_f32_16x16x4_f32(
            false, a, false, b, (short)0, c, false, false);

#pragma unroll
        for (int v = 0; v < 8; ++v) {
            const bool  m = ri[v] < rj;
            const float t = fmaxf(d[v], 0.0f);
            sum += m ? t * t : 0.0f;
            cnt += m ? 1u : 0u;
        }
    }

    // wave32 reduction
#pragma unroll
    for (int off = 16; off > 0; off >>= 1) {
        sum += __shfl_xor(sum, off, 32);
        cnt += (unsigned int)__shfl_xor((int)cnt, off, 32);
    }

    __shared__ float        s_sum[8];
    __shared__ unsigned int s_cnt[8];
    if (lane == 0) { s_sum[wave] = sum; s_cnt[wave] = cnt; }
    __syncthreads();

    if (threadIdx.x == 0) {
        float        bs = 0.0f;
        unsigned int bc = 0u;
#pragma unroll
        for (int w = 0; w < 8; ++w) { bs += s_sum[w]; bc += s_cnt[w]; }
        atomicAdd(ws_sum, bs);
        atomicAdd(ws_cnt, bc);
    }
}

__global__ void rankloss_final(const float* ws_sum, const unsigned int* ws_cnt,
                               float* out) {
    const unsigned int c = *ws_cnt;
    out[0] = *ws_sum / (float)(c ? c : 1u);
}

extern "C" void kernel_launch(void* const* d_in, const int* in_sizes, int n_in,
                              void* d_out, int out_size, void* d_ws, size_t ws_size,
                              hipStream_t stream) {
    const float* pred = (const float*)d_in[0];
    const int*   rank = (const int*)d_in[1];
    float*       out  = (float*)d_out;

    float*        ws_sum = (float*)d_ws;
    unsigned int* ws_cnt = (unsigned int*)((char*)d_ws + 64);

    rankloss_init<<<1, 1, 0, stream>>>(ws_sum, ws_cnt);

    dim3 grid(64, 8);          // 64 blocks x 8 waves = 512 i-tiles; y splits 512 j-tiles
    dim3 block(256);
    rankloss_wmma<<<grid, block, 0, stream>>>(pred, rank, ws_sum, ws_cnt);

    rankloss_final<<<1, 1, 0, stream>>>(ws_sum, ws_cnt, out);
}